// ADMM_LASSO_8486855377126
// MI455X (gfx1250) — compile-verified
//
#include <hip/hip_runtime.h>
#include <math.h>

typedef __attribute__((ext_vector_type(2))) float v2f;
typedef __attribute__((ext_vector_type(8))) float v8f;

#define RHO      1.0f
#define ORP      1.0f
#define MAX_ITER 500
#define NS_ITER  30
#define TOL_ABS  1e-4f
#define TOL_REL  1e-3f

// Block tile 64(M) x 128(N), K-step 16. 128 threads = 4 wave32.
// Waves arranged 2x2; each wave owns 32x64 = 2x4 WMMA 16x16 tiles.
#define BM 64
#define BN 128
#define BK 16

// ---------------------------------------------------------------------------
// M = A^T A + rho*I   (A is m x n row-major; output n x n)
// Double-buffered LDS, register-staged b128 global loads, WMMA f32 16x16x4.
// ---------------------------------------------------------------------------
__global__ __launch_bounds__(128)
void syrk_wmma(const float* __restrict__ A, float* __restrict__ Mout,
               int m, int n) {
  __shared__ float As[2][BK][BM];
  __shared__ float Bs[2][BK][BN];
  const int row0 = blockIdx.y * BM;
  const int col0 = blockIdx.x * BN;
  const int wid  = threadIdx.x >> 5;
  const int lane = threadIdx.x & 31;
  const int waveM = wid >> 1, waveN = wid & 1;
  // A-tile loader: 16x64 floats, thread -> (tkA, 4*tqA), two rows (tkA, tkA+8)
  const int tkA = threadIdx.x >> 4, tqA = threadIdx.x & 15;
  // B-tile loader: 16x128 floats, thread -> (tkB, 4*tqB), rows tkB+{0,4,8,12}
  const int tkB = threadIdx.x >> 5, tqB = threadIdx.x & 31;

  float4 pa[2], pb[4];
  v8f acc[2][4] = {};

  // prologue: stage tile k0=0
#pragma unroll
  for (int h = 0; h < 2; ++h)
    pa[h] = *(const float4*)&A[(size_t)(tkA + 8 * h) * n + row0 + 4 * tqA];
#pragma unroll
  for (int h = 0; h < 4; ++h)
    pb[h] = *(const float4*)&A[(size_t)(tkB + 4 * h) * n + col0 + 4 * tqB];
#pragma unroll
  for (int h = 0; h < 2; ++h) *(float4*)&As[0][tkA + 8 * h][4 * tqA] = pa[h];
#pragma unroll
  for (int h = 0; h < 4; ++h) *(float4*)&Bs[0][tkB + 4 * h][4 * tqB] = pb[h];
  __syncthreads();

  int buf = 0;
  for (int k0 = 0; k0 < m; k0 += BK) {
    const int kn = k0 + BK;
    if (kn < m) {
      if (kn + BK < m)  // hint tile after next -> global_prefetch_b8
        __builtin_prefetch(&A[(size_t)(kn + BK + tkA) * n + row0 + 4 * tqA], 0, 3);
#pragma unroll
      for (int h = 0; h < 2; ++h)
        pa[h] = *(const float4*)&A[(size_t)(kn + tkA + 8 * h) * n + row0 + 4 * tqA];
#pragma unroll
      for (int h = 0; h < 4; ++h)
        pb[h] = *(const float4*)&A[(size_t)(kn + tkB + 4 * h) * n + col0 + 4 * tqB];
    }
    // compute current buffer while next-tile loads are in flight
#pragma unroll
    for (int k4 = 0; k4 < BK; k4 += 4) {
      const int kb = k4 + ((lane >> 4) << 1);
      v2f a[2], b[4];
#pragma unroll
      for (int t = 0; t < 2; ++t) {
        int rr = waveM * 32 + t * 16 + (lane & 15);
        a[t].x = As[buf][kb][rr];  a[t].y = As[buf][kb + 1][rr];
      }
#pragma unroll
      for (int t = 0; t < 4; ++t) {
        int cc = waveN * 64 + t * 16 + (lane & 15);
        b[t].x = Bs[buf][kb][cc];  b[t].y = Bs[buf][kb + 1][cc];
      }
#pragma unroll
      for (int tm = 0; tm < 2; ++tm)
#pragma unroll
        for (int tn = 0; tn < 4; ++tn)
          acc[tm][tn] = __builtin_amdgcn_wmma_f32_16x16x4_f32(
              false, a[tm], false, b[tn], (short)0, acc[tm][tn], false, false);
    }
    if (kn < m) {
      const int nb = buf ^ 1;
#pragma unroll
      for (int h = 0; h < 2; ++h) *(float4*)&As[nb][tkA + 8 * h][4 * tqA] = pa[h];
#pragma unroll
      for (int h = 0; h < 4; ++h) *(float4*)&Bs[nb][tkB + 4 * h][4 * tqB] = pb[h];
      __syncthreads();
      buf = nb;
    }
  }
  // C/D layout: VGPR e -> M=e (lanes 0-15) / M=e+8 (lanes 16-31); N = lane&15
#pragma unroll
  for (int tm = 0; tm < 2; ++tm)
#pragma unroll
    for (int tn = 0; tn < 4; ++tn)
#pragma unroll
      for (int e = 0; e < 8; ++e) {
        int r = row0 + waveM * 32 + tm * 16 + e + ((lane >> 4) << 3);
        int c = col0 + waveN * 64 + tn * 16 + (lane & 15);
        float v = acc[tm][tn][e];
        if (r == c) v += RHO;
        Mout[(size_t)r * n + c] = v;
      }
}

// ---------------------------------------------------------------------------
// D = Am @ Bmod, Bmod = (MODE ? 2I - Bm : Bm).  All n x n row-major.
// A-tile staged transposed (k-major) with +1 padding to avoid bank conflicts.
// ---------------------------------------------------------------------------
template <int MODE>
__global__ __launch_bounds__(128)
void gemm_wmma(const float* __restrict__ Am, const float* __restrict__ Bm,
               float* __restrict__ Dm, int n) {
  __shared__ float As[2][BK][BM + 1];
  __shared__ float Bs[2][BK][BN];
  const int row0 = blockIdx.y * BM;
  const int col0 = blockIdx.x * BN;
  const int wid  = threadIdx.x >> 5;
  const int lane = threadIdx.x & 31;
  const int waveM = wid >> 1, waveN = wid & 1;
  // A loader: contiguous in k: thread -> (row ti/ti+32, k-quad 4*tqA)
  const int tqA = threadIdx.x & 3, tiA = threadIdx.x >> 2;
  // B loader: contiguous in j
  const int tkB = threadIdx.x >> 5, tqB = threadIdx.x & 31;

  float4 pa[2], pb[4];
  v8f acc[2][4] = {};

#pragma unroll
  for (int h = 0; h < 2; ++h)
    pa[h] = *(const float4*)&Am[(size_t)(row0 + tiA + 32 * h) * n + 4 * tqA];
#pragma unroll
  for (int h = 0; h < 4; ++h)
    pb[h] = *(const float4*)&Bm[(size_t)(tkB + 4 * h) * n + col0 + 4 * tqB];
  {
#pragma unroll
    for (int h = 0; h < 2; ++h) {
      As[0][4 * tqA + 0][tiA + 32 * h] = pa[h].x;
      As[0][4 * tqA + 1][tiA + 32 * h] = pa[h].y;
      As[0][4 * tqA + 2][tiA + 32 * h] = pa[h].z;
      As[0][4 * tqA + 3][tiA + 32 * h] = pa[h].w;
    }
#pragma unroll
    for (int h = 0; h < 4; ++h) {
      int kk = tkB + 4 * h;
      float vx = pb[h].x, vy = pb[h].y, vz = pb[h].z, vw = pb[h].w;
      if (MODE) {
        int c = col0 + 4 * tqB;
        vx = (kk == c + 0) ? 2.0f - vx : -vx;
        vy = (kk == c + 1) ? 2.0f - vy : -vy;
        vz = (kk == c + 2) ? 2.0f - vz : -vz;
        vw = (kk == c + 3) ? 2.0f - vw : -vw;
      }
      Bs[0][kk][4 * tqB + 0] = vx;  Bs[0][kk][4 * tqB + 1] = vy;
      Bs[0][kk][4 * tqB + 2] = vz;  Bs[0][kk][4 * tqB + 3] = vw;
    }
  }
  __syncthreads();

  int buf = 0;
  for (int k0 = 0; k0 < n; k0 += BK) {
    const int kn = k0 + BK;
    if (kn < n) {
      if (kn + BK < n)
        __builtin_prefetch(&Bm[(size_t)(kn + BK + tkB) * n + col0 + 4 * tqB], 0, 3);
#pragma unroll
      for (int h = 0; h < 2; ++h)
        pa[h] = *(const float4*)&Am[(size_t)(row0 + tiA + 32 * h) * n + kn + 4 * tqA];
#pragma unroll
      for (int h = 0; h < 4; ++h)
        pb[h] = *(const float4*)&Bm[(size_t)(kn + tkB + 4 * h) * n + col0 + 4 * tqB];
    }
#pragma unroll
    for (int k4 = 0; k4 < BK; k4 += 4) {
      const int kb = k4 + ((lane >> 4) << 1);
      v2f a[2], b[4];
#pragma unroll
      for (int t = 0; t < 2; ++t) {
        int rr = waveM * 32 + t * 16 + (lane & 15);
        a[t].x = As[buf][kb][rr];  a[t].y = As[buf][kb + 1][rr];
      }
#pragma unroll
      for (int t = 0; t < 4; ++t) {
        int cc = waveN * 64 + t * 16 + (lane & 15);
        b[t].x = Bs[buf][kb][cc];  b[t].y = Bs[buf][kb + 1][cc];
      }
#pragma unroll
      for (int tm = 0; tm < 2; ++tm)
#pragma unroll
        for (int tn = 0; tn < 4; ++tn)
          acc[tm][tn] = __builtin_amdgcn_wmma_f32_16x16x4_f32(
              false, a[tm], false, b[tn], (short)0, acc[tm][tn], false, false);
    }
    if (kn < n) {
      const int nb = buf ^ 1;
#pragma unroll
      for (int h = 0; h < 2; ++h) {
        As[nb][4 * tqA + 0][tiA + 32 * h] = pa[h].x;
        As[nb][4 * tqA + 1][tiA + 32 * h] = pa[h].y;
        As[nb][4 * tqA + 2][tiA + 32 * h] = pa[h].z;
        As[nb][4 * tqA + 3][tiA + 32 * h] = pa[h].w;
      }
#pragma unroll
      for (int h = 0; h < 4; ++h) {
        int kk = tkB + 4 * h;
        float vx = pb[h].x, vy = pb[h].y, vz = pb[h].z, vw = pb[h].w;
        if (MODE) {
          int c = col0 + 4 * tqB;
          int kg = kn + kk;
          vx = (kg == c + 0) ? 2.0f - vx : -vx;
          vy = (kg == c + 1) ? 2.0f - vy : -vy;
          vz = (kg == c + 2) ? 2.0f - vz : -vz;
          vw = (kg == c + 3) ? 2.0f - vw : -vw;
        }
        Bs[nb][kk][4 * tqB + 0] = vx;  Bs[nb][kk][4 * tqB + 1] = vy;
        Bs[nb][kk][4 * tqB + 2] = vz;  Bs[nb][kk][4 * tqB + 3] = vw;
      }
      __syncthreads();
      buf = nb;
    }
  }
#pragma unroll
  for (int tm = 0; tm < 2; ++tm)
#pragma unroll
    for (int tn = 0; tn < 4; ++tn)
#pragma unroll
      for (int e = 0; e < 8; ++e) {
        int r = row0 + waveM * 32 + tm * 16 + e + ((lane >> 4) << 3);
        int c = col0 + waveN * 64 + tn * 16 + (lane & 15);
        Dm[(size_t)r * n + c] = acc[tm][tn][e];
      }
}

// NOTE for MODE: first prologue uses k0=0 so global k == kk there; handled
// above since kn==0 is impossible (prologue uses literal k0=0 and kk==global k).

// ---------------------------------------------------------------------------
// ATb[i] = sum_k A[k][i] * b[k]   (coalesced across i)
// ---------------------------------------------------------------------------
__global__ void atb_kernel(const float* __restrict__ A, const float* __restrict__ b,
                           float* __restrict__ ATb, int m, int n) {
  int i = blockIdx.x * blockDim.x + threadIdx.x;
  if (i >= n) return;
  float s = 0.0f;
  for (int k = 0; k < m; ++k) s += A[(size_t)k * n + i] * b[k];
  ATb[i] = s;
}

// rs[row] = sum_c |M[row][c]|   (one wave32 per row, float4 streams)
__global__ __launch_bounds__(256)
void rowabssum(const float* __restrict__ M, float* __restrict__ rs, int n) {
  int wid = threadIdx.x >> 5, lane = threadIdx.x & 31;
  int row = blockIdx.x * 8 + wid;
  if (row >= n) return;
  const float* p = M + (size_t)row * n;
  float s = 0.0f;
  for (int c0 = lane * 4; c0 < n; c0 += 128) {
    float4 v = *(const float4*)(p + c0);
    s += fabsf(v.x) + fabsf(v.y) + fabsf(v.z) + fabsf(v.w);
  }
  for (int off = 16; off; off >>= 1) s += __shfl_down(s, off, 32);
  if (lane == 0) rs[row] = s;
}

// scale = 1 / (max_row_abs_sum)^2   (||M||_1 == ||M||_inf for symmetric M)
__global__ void maxscale(const float* __restrict__ rs, float* scalep, int n) {
  __shared__ float sm[256];
  float mx = 0.0f;
  for (int i = threadIdx.x; i < n; i += 256) mx = fmaxf(mx, rs[i]);
  sm[threadIdx.x] = mx;
  __syncthreads();
  for (int s = 128; s; s >>= 1) {
    if ((int)threadIdx.x < s) sm[threadIdx.x] = fmaxf(sm[threadIdx.x], sm[threadIdx.x + s]);
    __syncthreads();
  }
  if (threadIdx.x == 0) scalep[0] = 1.0f / (sm[0] * sm[0]);
}

__global__ void scalemat(const float* __restrict__ M, float* __restrict__ Y,
                         const float* __restrict__ scalep, size_t nn4) {
  size_t i = blockIdx.x * (size_t)blockDim.x + threadIdx.x;
  if (i >= nn4) return;
  float s = scalep[0];
  float4 v = ((const float4*)M)[i];
  v.x *= s; v.y *= s; v.z *= s; v.w *= s;
  ((float4*)Y)[i] = v;
}

__global__ void admm_init(float* x, float* z, float* y, float* acc, int* done, int n) {
  int i = blockIdx.x * blockDim.x + threadIdx.x;
  if (i < n) { x[i] = 0.0f; z[i] = 0.0f; y[i] = 0.0f; }
  if (i == 0) {
    for (int j = 0; j < 5; ++j) acc[j] = 0.0f;
    *done = 0;
  }
}

__global__ void vec_q(const float* __restrict__ ATb, const float* __restrict__ z,
                      const float* __restrict__ y, float* __restrict__ q,
                      const int* __restrict__ done, int n) {
  if (*done) return;
  int i = blockIdx.x * blockDim.x + threadIdx.x;
  if (i < n) q[i] = ATb[i] + RHO * (z[i] - y[i]);
}

// x = P @ q : one wave32 per row, L2-resident P (64MB < 192MB L2)
__global__ __launch_bounds__(256)
void pmatvec(const float* __restrict__ P, const float* __restrict__ q,
             float* __restrict__ x, const int* __restrict__ done, int n) {
  if (*done) return;
  int wid = threadIdx.x >> 5, lane = threadIdx.x & 31;
  int row = blockIdx.x * 8 + wid;
  if (row >= n) return;
  const float* p = P + (size_t)row * n;
  float s = 0.0f;
  for (int c0 = lane * 4; c0 < n; c0 += 128) {
    float4 pv = *(const float4*)(p + c0);
    float4 qv = *(const float4*)(q + c0);
    s += pv.x * qv.x + pv.y * qv.y + pv.z * qv.z + pv.w * qv.w;
  }
  for (int off = 16; off; off >>= 1) s += __shfl_down(s, off, 32);
  if (lane == 0) x[row] = s;
}

__global__ __launch_bounds__(256)
void admm_update(const float* __restrict__ x, float* __restrict__ z, float* __restrict__ y,
                 const float* __restrict__ alpha, float* acc,
                 const int* __restrict__ done, int n) {
  if (*done) return;
  int i = blockIdx.x * blockDim.x + threadIdx.x;
  float pri2 = 0, dual2 = 0, xn2 = 0, zn2 = 0, yn2 = 0;
  if (i < n) {
    float lam = alpha[0] / RHO;
    float xv = x[i], zo = z[i], yo = y[i];
    float xh = ORP * xv + (1.0f - ORP) * zo;
    float v  = xh + yo;
    float zv = copysignf(fmaxf(fabsf(v) - lam, 0.0f), v);
    float yv = yo + xh - zv;
    z[i] = zv; y[i] = yv;
    float dp = xv - zv;         pri2  = dp * dp;
    float dd = RHO * (zv - zo); dual2 = dd * dd;
    xn2 = xv * xv;  zn2 = zv * zv;
    float ry = RHO * yv;        yn2   = ry * ry;
  }
  for (int off = 16; off; off >>= 1) {
    pri2  += __shfl_down(pri2,  off, 32);
    dual2 += __shfl_down(dual2, off, 32);
    xn2   += __shfl_down(xn2,   off, 32);
    zn2   += __shfl_down(zn2,   off, 32);
    yn2   += __shfl_down(yn2,   off, 32);
  }
  if ((threadIdx.x & 31) == 0) {
    atomicAdd(&acc[0], pri2);  atomicAdd(&acc[1], dual2);
    atomicAdd(&acc[2], xn2);   atomicAdd(&acc[3], zn2);
    atomicAdd(&acc[4], yn2);
  }
}

__global__ void admm_finalize(float* acc, int* done, float sqrtn) {
  if (*done) return;  // acc already zero from previous finalize
  float pri = sqrtf(acc[0]), dual = sqrtf(acc[1]);
  float xn = sqrtf(acc[2]), zn = sqrtf(acc[3]), yn = sqrtf(acc[4]);
  float eps_pri  = sqrtn * TOL_ABS + TOL_REL * fmaxf(xn, zn);
  float eps_dual = sqrtn * TOL_ABS + TOL_REL * yn;
  if (pri < eps_pri && dual < eps_dual) *done = 1;
  acc[0] = acc[1] = acc[2] = acc[3] = acc[4] = 0.0f;
}

__global__ void copy_out(const float* __restrict__ x, float* __restrict__ out, int n) {
  int i = blockIdx.x * blockDim.x + threadIdx.x;
  if (i < n) out[i] = x[i];
}

// ---------------------------------------------------------------------------
extern "C" void kernel_launch(void* const* d_in, const int* in_sizes, int n_in,
                              void* d_out, int out_size, void* d_ws, size_t ws_size,
                              hipStream_t stream) {
  const float* A     = (const float*)d_in[0];
  const float* b     = (const float*)d_in[1];
  const float* alpha = (const float*)d_in[2];
  const int m = in_sizes[1];            // 2048
  const int n = in_sizes[0] / m;        // 4096
  const size_t nn = (size_t)n * n;

  size_t need = (4 * nn + 6 * (size_t)n + 16) * sizeof(float);
  if (ws_size < need) return;

  float* ws  = (float*)d_ws;
  float* Mm  = ws;              // ATA + rho I
  float* Y   = ws + nn;         // Newton-Schulz iterate
  float* T   = ws + 2 * nn;     // temp = M @ Y
  float* Y2  = ws + 3 * nn;     // ping-pong
  float* vec = ws + 4 * nn;
  float* ATb = vec;
  float* q   = vec + n;
  float* x   = vec + 2 * (size_t)n;
  float* z   = vec + 3 * (size_t)n;
  float* y   = vec + 4 * (size_t)n;
  float* rs  = vec + 5 * (size_t)n;
  float* acc = vec + 6 * (size_t)n;     // 5 norm accumulators
  float* scalep = acc + 5;
  int*   done   = (int*)(acc + 6);

  dim3 gTile(n / BN, n / BM);

  // 1) M = A^T A + rho I   (WMMA f32)
  syrk_wmma<<<gTile, 128, 0, stream>>>(A, Mm, m, n);
  // 2) ATb
  atb_kernel<<<(n + 255) / 256, 256, 0, stream>>>(A, b, ATb, m, n);
  // 3) Newton-Schulz init: Y0 = M / (||M||_1 * ||M||_inf)
  rowabssum<<<n / 8, 256, 0, stream>>>(Mm, rs, n);
  maxscale<<<1, 256, 0, stream>>>(rs, scalep, n);
  scalemat<<<(unsigned)((nn / 4 + 255) / 256), 256, 0, stream>>>(Mm, Y, scalep, nn / 4);
  // 4) Newton-Schulz: Y <- Y (2I - M Y)    (2 WMMA GEMMs / iter)
  float* Yc = Y;
  float* Yn = Y2;
  for (int it = 0; it < NS_ITER; ++it) {
    gemm_wmma<0><<<gTile, 128, 0, stream>>>(Mm, Yc, T, n);   // T = M @ Yc
    gemm_wmma<1><<<gTile, 128, 0, stream>>>(Yc, T, Yn, n);   // Yn = Yc @ (2I - T)
    float* t = Yc; Yc = Yn; Yn = t;
  }
  // 5) ADMM iterations with device-side convergence freeze
  admm_init<<<(n + 255) / 256, 256, 0, stream>>>(x, z, y, acc, done, n);
  float sqrtn = sqrtf((float)n);
  for (int it = 0; it < MAX_ITER; ++it) {
    vec_q<<<(n + 255) / 256, 256, 0, stream>>>(ATb, z, y, q, done, n);
    pmatvec<<<n / 8, 256, 0, stream>>>(Yc, q, x, done, n);
    admm_update<<<(n + 255) / 256, 256, 0, stream>>>(x, z, y, alpha, acc, done, n);
    admm_finalize<<<1, 1, 0, stream>>>(acc, done, sqrtn);
  }
  copy_out<<<(n + 255) / 256, 256, 0, stream>>>(x, (float*)d_out, n);
}